// GCN3_encoder_16226386444398
// MI455X (gfx1250) — compile-verified
//
#include <hip/hip_runtime.h>
#include <stdint.h>

typedef __attribute__((ext_vector_type(2))) float v2f;
typedef __attribute__((ext_vector_type(8))) float v8f;

#define NF 128          // F == H == 128
#define EPB 8           // edges per block in scatter kernel
#define ROWS 64         // rows per block in postact kernel

// ---------------- graph precompute (run once, reused by all 3 layers) --------

__global__ void k_init_deg(float* deg, int n) {
  int i = blockIdx.x * blockDim.x + threadIdx.x;
  if (i < n) deg[i] = 1.0f;                    // self-loop weight 1
}

__global__ void k_deg(const long long* __restrict__ dst,
                      const float* __restrict__ ew, float* deg, int e) {
  int i = blockIdx.x * blockDim.x + threadIdx.x;
  if (i < e) atomicAdd(&deg[dst[i]], ew[i]);
}

__global__ void k_dinv(const float* __restrict__ deg, float* dinv,
                       float* selfn, int n) {
  int i = blockIdx.x * blockDim.x + threadIdx.x;
  if (i < n) {
    float d = deg[i];
    float r = d > 0.0f ? rsqrtf(d) : 0.0f;
    dinv[i]  = r;
    selfn[i] = r * r;                          // self edge: dinv*1*dinv
  }
}

__global__ void k_edgenorm(const long long* __restrict__ src,
                           const long long* __restrict__ dst,
                           const float* __restrict__ ew,
                           const float* __restrict__ dinv,
                           float* __restrict__ enorm, int e) {
  int i = blockIdx.x * blockDim.x + threadIdx.x;
  if (i < e) enorm[i] = dinv[src[i]] * ew[i] * dinv[dst[i]];
}

// ---------------- GEMM: H = affine(X) @ W^T  (fp32 WMMA 16x16x4) -------------
// block = 256 threads = 8 waves; block handles 16 rows x 128 cols.
// wave w computes the 16x16 tile at cols [16w,16w+16).
// fp32 A 16x4 layout: lanes 0-15 hold K=k0,k0+1; lanes 16-31 hold K=k0+2,k0+3.
__global__ __launch_bounds__(256) void k_gemm(
    const float* __restrict__ X, const float* __restrict__ W,
    const float* __restrict__ scale, const float* __restrict__ shift,
    float* __restrict__ H, int n) {
  const int wave = threadIdx.x >> 5;
  const int lane = threadIdx.x & 31;
  const int half = lane >> 4;
  const int l    = lane & 15;
  const int m0   = blockIdx.x * 16;
  int m = m0 + l;
  if (m >= n) m = n - 1;                       // clamp reads (stores guarded)
  const int ncol = wave * 16 + l;              // B column == W row
  const float* __restrict__ xrow = X + (size_t)m * NF;
  const float* __restrict__ wrow = W + (size_t)ncol * NF;

  v8f acc = {};
  if (scale) {                                 // fused BN-apply from prev layer
    #pragma unroll 4
    for (int k0 = 0; k0 < NF; k0 += 4) {
      const int ka = k0 + half * 2;
      v2f a, b;
      a.x = xrow[ka]     * scale[ka]     + shift[ka];
      a.y = xrow[ka + 1] * scale[ka + 1] + shift[ka + 1];
      b.x = wrow[ka];
      b.y = wrow[ka + 1];
      acc = __builtin_amdgcn_wmma_f32_16x16x4_f32(
          false, a, false, b, (short)0, acc, false, false);
    }
  } else {
    #pragma unroll 4
    for (int k0 = 0; k0 < NF; k0 += 4) {
      const int ka = k0 + half * 2;
      v2f a, b;
      a.x = xrow[ka];
      a.y = xrow[ka + 1];
      b.x = wrow[ka];
      b.y = wrow[ka + 1];
      acc = __builtin_amdgcn_wmma_f32_16x16x4_f32(
          false, a, false, b, (short)0, acc, false, false);
    }
  }
  // D layout: VGPR r -> row m0 + r + half*8, col = wave*16 + l
  #pragma unroll
  for (int r = 0; r < 8; ++r) {
    const int row = m0 + r + half * 8;
    if (row < n) H[(size_t)row * NF + ncol] = acc[r];
  }
}

// ---------------- aggregation --------------------------------------------

__global__ void k_selfinit(const float* __restrict__ h,
                           const float* __restrict__ selfn,
                           float* __restrict__ agg, int total) {
  int i = blockIdx.x * blockDim.x + threadIdx.x;
  if (i < total) agg[i] = h[i] * selfn[i >> 7];
}

__global__ __launch_bounds__(128) void k_edge(
    const float* __restrict__ h, const long long* __restrict__ src,
    const long long* __restrict__ dst, const float* __restrict__ enorm,
    float* __restrict__ agg, int e) {
  const int f = threadIdx.x;
  const int base = blockIdx.x * EPB;
  #pragma unroll
  for (int i = 0; i < EPB; ++i) {
    const int eidx = base + i;
    if (eidx >= e) return;
    const long long s = src[eidx];
    const long long d = dst[eidx];
    const float w = enorm[eidx];
    atomicAdd(&agg[(size_t)d * NF + f], h[(size_t)s * NF + f] * w);
  }
}

// ---------------- bias + relu (in place) + BN stats ----------------------

__global__ void k_zero(float* p, int n) {
  int i = blockIdx.x * blockDim.x + threadIdx.x;
  if (i < n) p[i] = 0.0f;
}

__global__ __launch_bounds__(128) void k_postact(
    float* __restrict__ agg, const float* __restrict__ bias,
    float* __restrict__ stats, int n) {
  const int f = threadIdx.x;
  const float b = bias[f];
  float s = 0.0f, s2 = 0.0f;
  const int base = blockIdx.x * ROWS;
  for (int r = 0; r < ROWS; ++r) {
    const int row = base + r;
    if (row >= n) break;
    float v = agg[(size_t)row * NF + f] + b;
    v = v > 0.0f ? v : 0.0f;
    agg[(size_t)row * NF + f] = v;             // in-place relu
    s += v;
    s2 += v * v;
  }
  atomicAdd(&stats[f], s);
  atomicAdd(&stats[NF + f], s2);
}

__global__ void k_bnparams(const float* __restrict__ stats,
                           const float* __restrict__ g,
                           const float* __restrict__ be,
                           float* __restrict__ scsh, int n) {
  const int f = threadIdx.x;                   // 128 threads
  const float inv_n = 1.0f / (float)n;
  const float mean = stats[f] * inv_n;
  const float var  = stats[NF + f] * inv_n - mean * mean;
  const float sc = g[f] * rsqrtf(var + 1e-5f);
  scsh[f]      = sc;
  scsh[NF + f] = be[f] - mean * sc;
}

__global__ void k_affine_out(const float* __restrict__ y,
                             const float* __restrict__ scsh,
                             float* __restrict__ out, int total) {
  int i = blockIdx.x * blockDim.x + threadIdx.x;
  if (i < total) {
    const int f = i & (NF - 1);
    out[i] = y[i] * scsh[f] + scsh[NF + f];
  }
}

// ---------------- driver -------------------------------------------------

extern "C" void kernel_launch(void* const* d_in, const int* in_sizes, int n_in,
                              void* d_out, int out_size, void* d_ws, size_t ws_size,
                              hipStream_t stream) {
  const float*      x   = (const float*)d_in[0];
  const long long*  ei  = (const long long*)d_in[1];
  const float*      ew  = (const float*)d_in[2];
  const float* Wm[3] = {(const float*)d_in[3], (const float*)d_in[7],  (const float*)d_in[11]};
  const float* bb[3] = {(const float*)d_in[4], (const float*)d_in[8],  (const float*)d_in[12]};
  const float* gg[3] = {(const float*)d_in[5], (const float*)d_in[9],  (const float*)d_in[13]};
  const float* be[3] = {(const float*)d_in[6], (const float*)d_in[10], (const float*)d_in[14]};

  const int N = in_sizes[0] / NF;
  const int E = in_sizes[2];
  const long long* src = ei;
  const long long* dst = ei + E;
  const int total = N * NF;

  float* ws = (float*)d_ws;
  size_t off = 0;
  float* h     = ws + off; off += (size_t)N * NF;   // gemm output
  float* agg   = ws + off; off += (size_t)N * NF;   // aggregation / activations
  float* deg   = ws + off; off += N;
  float* dinv  = ws + off; off += N;
  float* selfn = ws + off; off += N;
  float* enorm = ws + off; off += E;
  float* stats = ws + off; off += 2 * NF;
  float* scsh  = ws + off; off += 2 * NF;

  // graph normalization: computed once, reused for all 3 layers
  k_init_deg<<<(N + 255) / 256, 256, 0, stream>>>(deg, N);
  k_deg<<<(E + 255) / 256, 256, 0, stream>>>(dst, ew, deg, E);
  k_dinv<<<(N + 255) / 256, 256, 0, stream>>>(deg, dinv, selfn, N);
  k_edgenorm<<<(E + 255) / 256, 256, 0, stream>>>(src, dst, ew, dinv, enorm, E);

  const int gemm_blocks = (N + 15) / 16;
  const float* in = x;
  const float* sc = nullptr;   // layer 1: identity affine on input
  const float* sh = nullptr;

  for (int L = 0; L < 3; ++L) {
    k_gemm<<<gemm_blocks, 256, 0, stream>>>(in, Wm[L], sc, sh, h, N);
    k_selfinit<<<(total + 255) / 256, 256, 0, stream>>>(h, selfn, agg, total);
    k_edge<<<(E + EPB - 1) / EPB, 128, 0, stream>>>(h, src, dst, enorm, agg, E);
    k_zero<<<1, 2 * NF, 0, stream>>>(stats, 2 * NF);
    k_postact<<<(N + ROWS - 1) / ROWS, 128, 0, stream>>>(agg, bb[L], stats, N);
    k_bnparams<<<1, NF, 0, stream>>>(stats, gg[L], be[L], scsh, N);
    in = agg;                  // next GEMM reads un-normalized activations
    sc = scsh;                 // ...and applies BN scale/shift on the fly
    sh = scsh + NF;
  }

  // final BN applied explicitly into d_out
  k_affine_out<<<(total + 255) / 256, 256, 0, stream>>>(agg, scsh, (float*)d_out, total);
}